// NewTTS_68891275428319
// MI455X (gfx1250) — compile-verified
//
#include <hip/hip_runtime.h>
#include <hip/hip_bf16.h>
#include <math.h>

// ---------------------------------------------------------------------------
// Problem constants (from reference)
// ---------------------------------------------------------------------------
#define BN     512      // batch N
#define UU     2048     // attention length U
#define VV     43       // cx feature dim V
#define HH     512      // hidden H
#define KK     10       // mixture components K
#define NCLS   512      // NCLASS
#define G3H    1536     // 3*H
#define CATD   555      // H + V (abk input dim)
#define X2D    44       // 1 + V
#define EPSC   1e-05f
#define WSEG   4        // u-segments for the cx einsum

typedef __attribute__((ext_vector_type(2))) float v2f;
typedef __attribute__((ext_vector_type(8))) float v8f;

// ---------------------------------------------------------------------------
// Kernel 1: abk = [eh | wt_1] @ Wh1p_w.T + b ; alpha/beta -> ws, kappa -> out
// One wave (32 lanes) per output scalar (N*3K = 15360 outputs).
// ---------------------------------------------------------------------------
__global__ void k_abk(const float* __restrict__ eh,      // (N,H)
                      const float* __restrict__ wt_1,    // (N,V)
                      const float* __restrict__ Ww,      // (3K, 555)
                      const float* __restrict__ Wb,      // (3K)
                      const float* __restrict__ kappa_1, // (N,K)
                      float* __restrict__ ws_alpha,      // (N,K)
                      float* __restrict__ ws_beta,       // (N,K)
                      float* __restrict__ out_kappa)     // (N,K)
{
    const int wid  = threadIdx.x >> 5;
    const int lane = threadIdx.x & 31;
    const int g    = blockIdx.x * 8 + wid;      // 0 .. 15359
    const int n    = g / (3 * KK);
    const int j    = g % (3 * KK);

    const float* __restrict__ wrow = Ww + j * CATD;
    const float* __restrict__ ehn  = eh + n * HH;
    const float* __restrict__ wtn  = wt_1 + n * VV;

    float s = 0.f;
    for (int i = lane; i < CATD; i += 32) {
        float x = (i < HH) ? ehn[i] : wtn[i - HH];
        s = fmaf(x, wrow[i], s);
    }
    // wave32 tree reduce
    #pragma unroll
    for (int off = 16; off > 0; off >>= 1)
        s += __shfl_xor(s, off, 32);

    if (lane == 0) {
        s += Wb[j];
        if (j < KK)              ws_alpha[n * KK + j]        = expf(s) + EPSC;
        else if (j < 2 * KK)     ws_beta [n * KK + (j - KK)] = expf(s) + EPSC;
        else out_kappa[n * KK + (j - 2 * KK)] =
                 kappa_1[n * KK + (j - 2 * KK)] + expf(s);
    }
}

// ---------------------------------------------------------------------------
// Kernel 2: phi[n][u] = sum_k alpha*exp(-beta*(kappa-u)^2)
// grid (U/256, N), block 256
// ---------------------------------------------------------------------------
__global__ void k_phi(const float* __restrict__ ws_alpha,
                      const float* __restrict__ ws_beta,
                      const float* __restrict__ out_kappa,
                      float* __restrict__ phi)           // (N,U)
{
    __shared__ float sa[KK], sb[KK], sk[KK];
    const int n = blockIdx.y;
    if (threadIdx.x < KK) {
        sa[threadIdx.x] = ws_alpha[n * KK + threadIdx.x];
        sb[threadIdx.x] = ws_beta [n * KK + threadIdx.x];
        sk[threadIdx.x] = out_kappa[n * KK + threadIdx.x];
    }
    __syncthreads();
    const int   u  = blockIdx.x * 256 + threadIdx.x;
    const float uf = (float)u;
    float s = 0.f;
    #pragma unroll
    for (int k = 0; k < KK; ++k) {
        float d = sk[k] - uf;
        s = fmaf(sa[k], __expf(-sb[k] * d * d), s);
    }
    phi[n * UU + u] = s;
}

// ---------------------------------------------------------------------------
// Kernel 3a: partial wt over a U-segment.
// grid (N, WSEG); block (64,4); phi chunk staged in LDS; contiguous cx stream.
// This is the HBM-bandwidth floor (reads 180 MB of cx total).
// ---------------------------------------------------------------------------
__global__ void k_wt_part(const float* __restrict__ cx,   // (N,U,V)
                          const float* __restrict__ phi,  // (N,U)
                          float* __restrict__ part)       // (N,WSEG,64)
{
    __shared__ float sphi[UU / WSEG];
    __shared__ float red[4][64];
    const int n   = blockIdx.x;
    const int seg = blockIdx.y;
    const int u0  = seg * (UU / WSEG);

    for (int i = threadIdx.y * 64 + threadIdx.x; i < UU / WSEG; i += 256)
        sphi[i] = phi[n * UU + u0 + i];
    __syncthreads();

    const int v = threadIdx.x;
    float acc = 0.f;
    if (v < VV) {
        const float* __restrict__ base = cx + (size_t)n * UU * VV + (size_t)u0 * VV + v;
        for (int u = threadIdx.y; u < UU / WSEG; u += 4)
            acc = fmaf(base[(size_t)u * VV], sphi[u], acc);
    }
    red[threadIdx.y][threadIdx.x] = acc;
    __syncthreads();
    if (threadIdx.y == 0)
        part[((size_t)n * WSEG + seg) * 64 + v] = red[0][v] + red[1][v] + red[2][v] + red[3][v];
}

// ---------------------------------------------------------------------------
// Kernel 3b: reduce partials -> wt, and build x2 = [xt | wt]
// ---------------------------------------------------------------------------
__global__ void k_wt_reduce(const float* __restrict__ part, // (N,WSEG,64)
                            const float* __restrict__ xt,   // (N)
                            float* __restrict__ wt,         // (N,V)
                            float* __restrict__ x2)         // (N,44)
{
    const int idx = blockIdx.x * 256 + threadIdx.x;
    if (idx >= BN * VV) return;
    const int n = idx / VV;
    const int v = idx % VV;
    float s = 0.f;
    #pragma unroll
    for (int sgi = 0; sgi < WSEG; ++sgi)
        s += part[((size_t)n * WSEG + sgi) * 64 + v];
    wt[n * VV + v] = s;
    x2[n * X2D + 1 + v] = s;
    if (v == 0) x2[n * X2D] = xt[n];
}

// ---------------------------------------------------------------------------
// WMMA f32 GEMM:  C[N x O] = A[N x K] @ W[O x K]^T + bias[O]
// One wave computes a 16 x (16*NT) output strip via V_WMMA_F32_16X16X4_F32,
// reusing the A fragment across NT column tiles (1 A-load + NT W-loads per
// NT WMMAs).  Software-pipelined one k-step ahead so the XDL pipe overlaps
// the L2 latency of the next fragment loads instead of waiting loadcnt==0
// before every WMMA.
//
// A-operand layout (ISA 7.12.2, 32-bit A 16x4): lane m = lane&15 holds row
// M=m; the VGPR pair holds K = (lane>>4)*2 and +1. B-operand (4x16, K x N)
// mirrors it: lane&15 is column N, VGPR pair holds the matching K pair — both
// sides are one aligned float2 load per k-step. C/D: VGPR r -> row
// rowTile*16 + r + (lane>>4)*8, col = colTile*16 + (lane&15).
// Full EXEC (no divergence) as WMMA requires.
// ---------------------------------------------------------------------------
template <int NT>
__global__ void k_gemm_wmma(const float* __restrict__ A,
                            const float* __restrict__ W,
                            const float* __restrict__ bias,
                            float* __restrict__ C,
                            int Kdim, int Odim)
{
    const int wid       = threadIdx.x >> 5;
    const int lane      = threadIdx.x & 31;
    const int t         = blockIdx.x * 8 + wid;       // strip id
    const int colGroups = Odim / (16 * NT);
    const int rowTile   = t / colGroups;
    const int colGroup  = t % colGroups;

    const int m  = lane & 15;
    const int ko = (lane >> 4) << 1;                  // 0 or 2

    const float* __restrict__ Ar = A + (size_t)(rowTile * 16 + m) * Kdim + ko;
    const float* Wr[NT];
    #pragma unroll
    for (int j = 0; j < NT; ++j)
        Wr[j] = W + (size_t)((colGroup * NT + j) * 16 + m) * Kdim + ko;

    v8f c[NT];
    #pragma unroll
    for (int j = 0; j < NT; ++j) c[j] = (v8f){};

    // prologue: fragments for k = 0
    float2 av = *(const float2*)(Ar);
    float2 wv[NT];
    #pragma unroll
    for (int j = 0; j < NT; ++j) wv[j] = *(const float2*)(Wr[j]);

    // steady state: issue k+4 loads, then WMMA on k's fragments
    for (int k = 0; k + 4 < Kdim; k += 4) {
        const float2 av_n = *(const float2*)(Ar + k + 4);
        float2 wv_n[NT];
        #pragma unroll
        for (int j = 0; j < NT; ++j) wv_n[j] = *(const float2*)(Wr[j] + k + 4);

        v2f a; a[0] = av.x; a[1] = av.y;
        #pragma unroll
        for (int j = 0; j < NT; ++j) {
            v2f b; b[0] = wv[j].x; b[1] = wv[j].y;
            // 8 args: (neg_a, A, neg_b, B, c_mod, C, reuse_a, reuse_b)
            c[j] = __builtin_amdgcn_wmma_f32_16x16x4_f32(
                       false, a, false, b, (short)0, c[j], false, false);
        }
        av = av_n;
        #pragma unroll
        for (int j = 0; j < NT; ++j) wv[j] = wv_n[j];
    }

    // epilogue WMMA for k = Kdim-4
    {
        v2f a; a[0] = av.x; a[1] = av.y;
        #pragma unroll
        for (int j = 0; j < NT; ++j) {
            v2f b; b[0] = wv[j].x; b[1] = wv[j].y;
            c[j] = __builtin_amdgcn_wmma_f32_16x16x4_f32(
                       false, a, false, b, (short)0, c[j], false, false);
        }
    }

    const int rbase = rowTile * 16 + ((lane >> 4) << 3);
    #pragma unroll
    for (int j = 0; j < NT; ++j) {
        const int   col = (colGroup * NT + j) * 16 + m;
        const float bv  = bias[col];
        #pragma unroll
        for (int r = 0; r < 8; ++r)
            C[(size_t)(rbase + r) * Odim + col] = c[j][r] + bv;
    }
}

// ---------------------------------------------------------------------------
// GRU gate combine: h = (1-z)*tanh(gin + r*ghn) + z*hprev
// Also writes act = relu(h [+ res]) for the next GEMM's A operand.
// ---------------------------------------------------------------------------
__global__ void k_gru_combine(const float* __restrict__ gi,   // (N,3H)
                              const float* __restrict__ gh,   // (N,3H)
                              const float* __restrict__ hprev,// (N,H)
                              const float* __restrict__ res,  // (N,H) or null
                              float* __restrict__ hOut,       // (N,H)
                              float* __restrict__ actOut)     // (N,H)
{
    const int idx = blockIdx.x * 256 + threadIdx.x;  // N*H threads exactly
    const int n = idx >> 9;          // /512
    const int j = idx & 511;
    const int b = n * G3H + j;

    const float gir = gi[b], giz = gi[b + HH], gin = gi[b + 2 * HH];
    const float ghr = gh[b], ghz = gh[b + HH], ghn = gh[b + 2 * HH];

    const float r  = 1.f / (1.f + expf(-(gir + ghr)));
    const float z  = 1.f / (1.f + expf(-(giz + ghz)));
    const float nn = tanhf(gin + r * ghn);
    const float h  = (1.f - z) * nn + z * hprev[idx];

    hOut[idx] = h;
    const float o = h + (res ? res[idx] : 0.f);
    actOut[idx] = fmaxf(o, 0.f);
}

// ---------------------------------------------------------------------------
// Launch
// ---------------------------------------------------------------------------
extern "C" void kernel_launch(void* const* d_in, const int* in_sizes, int n_in,
                              void* d_out, int out_size, void* d_ws, size_t ws_size,
                              hipStream_t stream)
{
    const float* xt       = (const float*)d_in[0];
    const float* cx       = (const float*)d_in[1];
    const float* enc_h    = (const float*)d_in[2];   // (1,N,H) -> flat (N,H)
    const float* wt_1     = (const float*)d_in[3];
    const float* h2t_1    = (const float*)d_in[4];   // (1,N,H)
    const float* h3t_1    = (const float*)d_in[5];
    const float* kappa_1  = (const float*)d_in[6];
    const float* Wh1p_w   = (const float*)d_in[7];
    const float* Wh1p_b   = (const float*)d_in[8];
    const float* rnn2_wih = (const float*)d_in[9];
    const float* rnn2_whh = (const float*)d_in[10];
    const float* rnn2_bih = (const float*)d_in[11];
    const float* rnn2_bhh = (const float*)d_in[12];
    const float* rnn3_wih = (const float*)d_in[13];
    const float* rnn3_whh = (const float*)d_in[14];
    const float* rnn3_bih = (const float*)d_in[15];
    const float* rnn3_bhh = (const float*)d_in[16];
    const float* Y_w      = (const float*)d_in[17];
    const float* Y_b      = (const float*)d_in[18];

    // Output layout: dist | h2 | h3 | phi | wt | kappa (flat, fp32)
    float* out_dist  = (float*)d_out;                      // 512*512
    float* out_h2    = out_dist + BN * NCLS;               // 512*512
    float* out_h3    = out_h2   + BN * HH;                 // 512*512
    float* out_phi   = out_h3   + BN * HH;                 // 512*2048
    float* out_wt    = out_phi  + BN * UU;                 // 512*43
    float* out_kappa = out_wt   + BN * VV;                 // 512*10

    // Workspace layout (floats); gi/gh buffers reused across both GRUs.
    float* ws        = (float*)d_ws;
    float* ws_alpha  = ws;                    //   5120
    float* ws_beta   = ws + 5120;             //   5120
    float* ws_x2     = ws + 10240;            //  22528
    float* ws_gi     = ws + 32768;            // 786432
    float* ws_gh     = ws + 819200;           // 786432
    float* ws_act    = ws + 1605632;          // 262144  (relu(h2), then relu(h2+h3))
    float* ws_wtpart = ws + 1867776;          // 512*WSEG*64 = 131072
    // total: 1998848 floats = ~8.0 MB

    // 1) attention mixture params
    k_abk<<<dim3((BN * 3 * KK) / 8), 256, 0, stream>>>(
        enc_h, wt_1, Wh1p_w, Wh1p_b, kappa_1, ws_alpha, ws_beta, out_kappa);

    // 2) phi window
    k_phi<<<dim3(UU / 256, BN), 256, 0, stream>>>(
        ws_alpha, ws_beta, out_kappa, out_phi);

    // 3) wt = einsum('nuv,nu->nv')   [HBM-bound: 180 MB of cx]
    k_wt_part<<<dim3(BN, WSEG), dim3(64, 4), 0, stream>>>(cx, out_phi, ws_wtpart);
    k_wt_reduce<<<dim3((BN * VV + 255) / 256), 256, 0, stream>>>(
        ws_wtpart, xt, out_wt, ws_x2);

    // 4) GRU2:  gi = x2 @ wih.T + bih ;  gh = h2t_1 @ whh.T + bhh
    {
        const int strips = (BN / 16) * (G3H / 32);         // 1536
        k_gemm_wmma<2><<<dim3(strips / 8), 256, 0, stream>>>(
            ws_x2, rnn2_wih, rnn2_bih, ws_gi, X2D, G3H);
        k_gemm_wmma<2><<<dim3(strips / 8), 256, 0, stream>>>(
            h2t_1, rnn2_whh, rnn2_bhh, ws_gh, HH, G3H);
        k_gru_combine<<<dim3((BN * HH) / 256), 256, 0, stream>>>(
            ws_gi, ws_gh, h2t_1, nullptr, out_h2, ws_act);
    }

    // 5) GRU3:  gi = relu(h2) @ wih.T + bih ;  gh = h3t_1 @ whh.T + bhh
    {
        const int strips = (BN / 16) * (G3H / 32);
        k_gemm_wmma<2><<<dim3(strips / 8), 256, 0, stream>>>(
            ws_act, rnn3_wih, rnn3_bih, ws_gi, HH, G3H);
        k_gemm_wmma<2><<<dim3(strips / 8), 256, 0, stream>>>(
            h3t_1, rnn3_whh, rnn3_bhh, ws_gh, HH, G3H);
        // h3 out; act = relu(h3 + h2)
        k_gru_combine<<<dim3((BN * HH) / 256), 256, 0, stream>>>(
            ws_gi, ws_gh, h3t_1, out_h2, out_h3, ws_act);
    }

    // 6) dist = relu(out) @ Y_w.T + Y_b
    {
        const int strips = (BN / 16) * (NCLS / 32);        // 512
        k_gemm_wmma<2><<<dim3(strips / 8), 256, 0, stream>>>(
            ws_act, Y_w, Y_b, out_dist, HH, NCLS);
    }
}